// RotatedIoULoss_19378892439684
// MI455X (gfx1250) — compile-verified
//
#include <hip/hip_runtime.h>
#include <math.h>

typedef float v2f __attribute__((ext_vector_type(2)));
typedef float v8f __attribute__((ext_vector_type(8)));

#define RIOU_EPS 1e-6f

__global__ __launch_bounds__(256) void riou_partial_kernel(
    const float* __restrict__ pred,
    const float* __restrict__ target,
    const float* __restrict__ weight,
    float* __restrict__ partial,
    int n)
{
  const int tid = threadIdx.x;
  const long long gid = (long long)blockIdx.x * 256 + tid;
  const int j = (gid < (long long)n) ? (int)gid : (n - 1);

  const float px = pred[j * 5 + 0];
  const float py = pred[j * 5 + 1];
  const float pw = pred[j * 5 + 2];
  const float ph = pred[j * 5 + 3];
  const float pa = pred[j * 5 + 4];
  const float qx = target[j * 5 + 0];
  const float qy = target[j * 5 + 1];
  const float qw = target[j * 5 + 2];
  const float qh = target[j * 5 + 3];
  const float qa = target[j * 5 + 4];
  const float wgt = weight[j];

  const float TXc[4] = {0.5f, -0.5f, -0.5f, 0.5f};
  const float TYc[4] = {0.5f, 0.5f, -0.5f, -0.5f};

  float c1x[4], c1y[4], c2x[4], c2y[4];
  const float sa = sinf(pa), ca = cosf(pa);
  const float sb = sinf(qa), cb = cosf(qa);
#pragma unroll
  for (int k = 0; k < 4; ++k) {
    const float ax = TXc[k] * pw, ay = TYc[k] * ph;
    c1x[k] = ax * ca - ay * sa + px;
    c1y[k] = ax * sa + ay * ca + py;
    const float bx = TXc[k] * qw, by = TYc[k] * qh;
    c2x[k] = bx * cb - by * sb + qx;
    c2y[k] = bx * sb + by * cb + qy;
  }

  float vx[24], vy[24];
  bool vm[24];

  // c1 corners inside c2  -> slots 0..3
  {
    const float abx = c2x[1] - c2x[0], aby = c2y[1] - c2y[0];
    const float adx = c2x[3] - c2x[0], ady = c2y[3] - c2y[0];
    const float nab = abx * abx + aby * aby;
    const float nad = adx * adx + ady * ady;
#pragma unroll
    for (int k = 0; k < 4; ++k) {
      const float amx = c1x[k] - c2x[0], amy = c1y[k] - c2y[0];
      const float pab = amx * abx + amy * aby;
      const float pad = amx * adx + amy * ady;
      vx[k] = c1x[k]; vy[k] = c1y[k];
      vm[k] = (pab > 0.f) & (pab < nab) & (pad > 0.f) & (pad < nad);
    }
  }
  // c2 corners inside c1  -> slots 4..7
  {
    const float abx = c1x[1] - c1x[0], aby = c1y[1] - c1y[0];
    const float adx = c1x[3] - c1x[0], ady = c1y[3] - c1y[0];
    const float nab = abx * abx + aby * aby;
    const float nad = adx * adx + ady * ady;
#pragma unroll
    for (int k = 0; k < 4; ++k) {
      const float amx = c2x[k] - c1x[0], amy = c2y[k] - c1y[0];
      const float pab = amx * abx + amy * aby;
      const float pad = amx * adx + amy * ady;
      vx[4 + k] = c2x[k]; vy[4 + k] = c2y[k];
      vm[4 + k] = (pab > 0.f) & (pab < nab) & (pad > 0.f) & (pad < nad);
    }
  }
  // edge-edge intersections -> slots 8..23 (index 8 + i*4 + jj, matching reshape order)
#pragma unroll
  for (int i = 0; i < 4; ++i) {
    const float p1x = c1x[i], p1y = c1y[i];
    const float d1x = c1x[(i + 1) & 3] - p1x, d1y = c1y[(i + 1) & 3] - p1y;
#pragma unroll
    for (int jj = 0; jj < 4; ++jj) {
      const float p2x = c2x[jj], p2y = c2y[jj];
      const float d2x = c2x[(jj + 1) & 3] - p2x, d2y = c2y[(jj + 1) & 3] - p2y;
      const float den = d1x * d2y - d1y * d2x;
      const float dpx = p2x - p1x, dpy = p2y - p1y;
      const float safe = (den == 0.f) ? 1.f : den;
      const float t = (dpx * d2y - dpy * d2x) / safe;
      const float u = (dpx * d1y - dpy * d1x) / safe;
      const int idx = 8 + i * 4 + jj;
      vx[idx] = p1x + t * d1x;
      vy[idx] = p1y + t * d1y;
      vm[idx] = (den != 0.f) & (t > 0.f) & (t < 1.f) & (u > 0.f) & (u < 1.f);
    }
  }

  int num = 0;
  float sxs = 0.f, sys = 0.f;
#pragma unroll
  for (int k = 0; k < 24; ++k) {
    if (vm[k]) { num++; sxs += vx[k]; sys += vy[k]; }
  }
  const float denom = (float)(num > 1 ? num : 1);
  const float cx0 = sxs / denom, cy0 = sys / denom;

  float ang[24];
#pragma unroll
  for (int k = 0; k < 24; ++k) {
    ang[k] = vm[k] ? atan2f(vy[k] - cy0, vx[k] - cx0) : 1.0e6f;
  }

  // Stable-rank compaction (equivalent to jnp stable argsort by (angle, index)):
  // valid vertices land compactly in sorted slots [0, num).
  float sx[24], sy[24];
  for (int k = 0; k < 24; ++k) {
    if (vm[k]) {
      int r = 0;
      for (int q = 0; q < 24; ++q) {
        const bool before = vm[q] && ((q < k) ? (ang[q] <= ang[k]) : (ang[q] < ang[k]));
        r += before ? 1 : 0;
      }
      sx[r] = vx[k];
      sy[r] = vy[k];
    }
  }

  float area2 = 0.f;
  for (int r = 0; r < num; ++r) {
    const int rn = (r + 1 == num) ? 0 : r + 1;
    area2 += sx[r] * sy[rn] - sy[r] * sx[rn];
  }
  const float inter = (num >= 3) ? 0.5f * fabsf(area2) : 0.f;

  const float a1 = pw * ph, a2 = qw * qh;
  float iou = inter / (a1 + a2 - inter);
  iou = fmaxf(iou, RIOU_EPS);
  float loss = -logf(iou) * wgt;
  if (gid >= (long long)n) loss = 0.f;   // keep lane active; contribute zero

  // ---- wave32 reduction through the matrix pipe: D = A(16x4) * ones(4x16) + C ----
  // A layout (32-bit, 16x4): lanes 0-15 hold A[M=lane][K=0..1], lanes 16-31 A[M=lane-16][K=2..3].
  // With A = {loss, 0}, rowsum_m = loss_m + loss_(m+16); D[m][n] = rowsum_m for all n.
  v2f amat; amat[0] = loss; amat[1] = 0.f;
  v2f bmat; bmat[0] = 1.f;  bmat[1] = 1.f;
  v8f cacc = {};
  cacc = __builtin_amdgcn_wmma_f32_16x16x4_f32(
      /*neg_a=*/false, amat, /*neg_b=*/false, bmat,
      /*c_mod=*/(short)0, cacc, /*reuse_a=*/false, /*reuse_b=*/false);
  // lane in 0-15: D regs = rows 0..7 (col = lane); lane in 16-31: rows 8..15.
  const float s8 = cacc[0] + cacc[1] + cacc[2] + cacc[3] +
                   cacc[4] + cacc[5] + cacc[6] + cacc[7];
  const float waveSum = __shfl(s8, 0, 32) + __shfl(s8, 16, 32);

  __shared__ float wsum[8];
  const int wid = tid >> 5;
  const int lane = tid & 31;
  if (lane == 0) wsum[wid] = waveSum;
  __syncthreads();
  if (tid == 0) {
    float b = 0.f;
#pragma unroll
    for (int w = 0; w < 8; ++w) b += wsum[w];
    partial[blockIdx.x] = b;
  }
}

__global__ __launch_bounds__(256) void riou_finalize_kernel(
    const float* __restrict__ partial, int nb, float* __restrict__ out, float inv_n)
{
  __shared__ float sm[256];
  float s = 0.f;
  for (int i = threadIdx.x; i < nb; i += 256) s += partial[i];
  sm[threadIdx.x] = s;
  __syncthreads();
#pragma unroll
  for (int off = 128; off > 0; off >>= 1) {
    if (threadIdx.x < (unsigned)off) sm[threadIdx.x] += sm[threadIdx.x + off];
    __syncthreads();
  }
  if (threadIdx.x == 0) out[0] = sm[0] * inv_n;   // LOSS_WEIGHT == 1.0
}

extern "C" void kernel_launch(void* const* d_in, const int* in_sizes, int n_in,
                              void* d_out, int out_size, void* d_ws, size_t ws_size,
                              hipStream_t stream) {
  (void)n_in; (void)out_size; (void)ws_size;
  const float* pred   = (const float*)d_in[0];
  const float* target = (const float*)d_in[1];
  const float* weight = (const float*)d_in[2];
  const int n = in_sizes[2];                 // weight has one element per box
  const int nb = (n + 255) / 256;
  float* partial = (float*)d_ws;

  riou_partial_kernel<<<nb, 256, 0, stream>>>(pred, target, weight, partial, n);
  riou_finalize_kernel<<<1, 256, 0, stream>>>(partial, nb, (float*)d_out, 1.0f / (float)n);
}